// MixDensityLoss_74071005987499
// MI455X (gfx1250) — compile-verified
//
#include <hip/hip_runtime.h>
#include <math.h>

#define B_ 16
#define T_ 800
#define L_ 160
#define D_ 80
#define KP 164                     // augmented K: D (mel^2) + D (mel) + 1 (bias) + 3 pad
#define KP2 (KP / 2)               // K-pair count (82)
#define LOG2PI 1.8378770664093453f
#define NEG_INF_F (-1e20f)

typedef float v2f __attribute__((ext_vector_type(2)));
typedef float v8f __attribute__((ext_vector_type(8)));

// ---------------------------------------------------------------------------
// Build pair-interleaved B-matrix [B, KP2, L, 2]: element (b,k,l) lives at
// ((b*KP2 + k/2)*L + l)*2 + (k&1).  k < D: inv_var; D <= k < 2D: -2*mu*iv;
// k == 2D: per-l bias = q_mu + sum(log_sigma) + D*log(2pi); else 0.
// Thread index runs (pair,l) fastest -> fully coalesced stores.
// ---------------------------------------------------------------------------
__global__ void prep_B_kernel(const float* __restrict__ mu_sigma,
                              float* __restrict__ Bmat) {
    int idx = blockIdx.x * blockDim.x + threadIdx.x;
    const int total = B_ * KP * L_;
    if (idx >= total) return;
    int p  = idx & 1;
    int l  = (idx >> 1) % L_;
    int kp = ((idx >> 1) / L_) % KP2;
    int b  = (idx >> 1) / (L_ * KP2);
    int k  = kp * 2 + p;

    const float* ms = mu_sigma + (size_t)(b * L_ + l) * (2 * D_);
    float val = 0.0f;
    if (k < D_) {
        float lsig = tanhf(ms[D_ + k]) * 4.0f;
        val = expf(-2.0f * lsig);
    } else if (k < 2 * D_) {
        int d = k - D_;
        float mu   = 4.0f / (1.0f + expf(-ms[d]));
        float lsig = tanhf(ms[D_ + d]) * 4.0f;
        val = -2.0f * mu * expf(-2.0f * lsig);
    } else if (k == 2 * D_) {
        float acc = (float)D_ * LOG2PI;
        for (int d = 0; d < D_; ++d) {
            float mu   = 4.0f / (1.0f + expf(-ms[d]));
            float lsig = tanhf(ms[D_ + d]) * 4.0f;
            float iv   = expf(-2.0f * lsig);
            acc += mu * mu * iv + lsig;
        }
        val = acc;
    }
    Bmat[idx] = val;
}

// ---------------------------------------------------------------------------
// Build pair-interleaved A^T [B, KP2, T, 2]: element (b,k,t) at
// ((b*KP2 + k/2)*T + t)*2 + (k&1).  k < D: mel^2; D<=k<2D: mel;
// k == 2D: 1.0 (bias lane); else 0.
// ---------------------------------------------------------------------------
__global__ void prep_At_kernel(const float* __restrict__ mel,
                               float* __restrict__ At) {
    int idx = blockIdx.x * blockDim.x + threadIdx.x;
    const int total = B_ * KP * T_;
    if (idx >= total) return;
    int p  = idx & 1;
    int t  = (idx >> 1) % T_;
    int kp = ((idx >> 1) / T_) % KP2;
    int b  = (idx >> 1) / (T_ * KP2);
    int k  = kp * 2 + p;

    float val = 0.0f;
    if (k < D_) {
        float m = mel[(size_t)(b * T_ + t) * D_ + k];
        val = m * m;
    } else if (k < 2 * D_) {
        val = mel[(size_t)(b * T_ + t) * D_ + (k - D_)];
    } else if (k == 2 * D_) {
        val = 1.0f;
    }
    At[idx] = val;
}

// ---------------------------------------------------------------------------
// One wave per 16(M=t) x 32(N=l) output tile: f32 WMMA over K'=164.
// Per K=4 step: one b64 A-fragment load feeds TWO wmmas (two N sub-tiles).
// Epilogue: log_prob = -0.5 * acc.
//
// f32 16x16x4 fragment layouts (ISA 7.12.2):
//   A (16x4):  lanes 0-15 -> M=lane, v={K0,K1};  lanes 16-31 -> {K2,K3}
//   B (4x16):  lanes 0-15 -> N=lane, v={K0,K1};  lanes 16-31 -> {K2,K3}
//   C (16x16): vgpr r: lanes 0-15 -> M=r, N=lane; lanes 16-31 -> M=8+r
// ---------------------------------------------------------------------------
__global__ void __launch_bounds__(32)
gemm_logprob_kernel(const float* __restrict__ At,
                    const float* __restrict__ Bmat,
                    float* __restrict__ logp) {
    const int tTile = blockIdx.x;      // T/16 = 50
    const int lTile = blockIdx.y;      // L/32 = 5
    const int b     = blockIdx.z;
    const int lane  = threadIdx.x;
    const int half  = lane >> 4;       // 0: K{0,1}, 1: K{2,3}
    const int sub   = lane & 15;

    const int t  = tTile * 16 + sub;        // A row for this lane
    const int n0 = lTile * 32 + sub;        // B col, first sub-tile
    const int n1 = n0 + 16;                 // B col, second sub-tile

    const v2f* Ab = (const v2f*)At   + (size_t)b * KP2 * T_;
    const v2f* Bb = (const v2f*)Bmat + (size_t)b * KP2 * L_;

    v8f acc0 = {};
    v8f acc1 = {};
    for (int k0 = 0; k0 < KP2; k0 += 2) {   // each step covers K=4
        const int kp = k0 + half;
        v2f a  = Ab[(size_t)kp * T_ + t];
        v2f b0 = Bb[(size_t)kp * L_ + n0];
        v2f b1 = Bb[(size_t)kp * L_ + n1];
        acc0 = __builtin_amdgcn_wmma_f32_16x16x4_f32(
            false, a, false, b0, (short)0, acc0, false, false);
        acc1 = __builtin_amdgcn_wmma_f32_16x16x4_f32(
            false, a, false, b1, (short)0, acc1, false, false);
    }

    float* out = logp + (size_t)b * T_ * L_;
    const int mbase = half * 8;
#pragma unroll
    for (int r = 0; r < 8; ++r) {
        const int tt = tTile * 16 + mbase + r;
        out[(size_t)tt * L_ + n0] = -0.5f * acc0[r];
        out[(size_t)tt * L_ + n1] = -0.5f * acc1[r];
    }
}

// ---------------------------------------------------------------------------
// Forward-algorithm scan: one wave per batch, 5 contiguous l-states per lane
// held in registers; one __shfl_up per step for the alpha[l-1] shift.
// mel_lengths[0]==T by construction, so max_mel==T and no row masking needed.
// ---------------------------------------------------------------------------
__global__ void __launch_bounds__(32)
scan_alphas_kernel(const float* __restrict__ logp,
                   float* __restrict__ alphas) {
    const int b    = blockIdx.x;
    const int lane = threadIdx.x;
    const float* lp = logp   + (size_t)b * T_ * L_;
    float*      out = alphas + (size_t)b * T_ * L_;
    const int l0 = lane * 5;

    float alpha[5];
#pragma unroll
    for (int j = 0; j < 5; ++j) {
        float v = (l0 + j == 0) ? lp[0] : NEG_INF_F;
        alpha[j] = v;
        out[l0 + j] = v;
    }

    for (int t = 1; t < T_; ++t) {
        float prev = __shfl_up(alpha[4], 1, 32);
        if (lane == 0) prev = NEG_INF_F;
        float na[5];
#pragma unroll
        for (int j = 0; j < 5; ++j) {
            float sh = (j == 0) ? prev : alpha[j - 1];
            float mx = fmaxf(alpha[j], sh);
            float df = fminf(alpha[j], sh) - mx;
            float lae = mx + log1pf(expf(df));
            na[j] = lae + lp[(size_t)t * L_ + l0 + j];
        }
#pragma unroll
        for (int j = 0; j < 5; ++j) {
            alpha[j] = na[j];
            out[(size_t)t * L_ + l0 + j] = na[j];
        }
    }
}

// ---------------------------------------------------------------------------
// loss = -mean_b alphas[b, mel_len[b]-1, char_len[b]-1]
// ---------------------------------------------------------------------------
__global__ void __launch_bounds__(32)
loss_kernel(const float* __restrict__ alphas,
            const int* __restrict__ mel_len,
            const int* __restrict__ char_len,
            float* __restrict__ loss) {
    const int lane = threadIdx.x;
    float v = 0.0f;
    if (lane < B_) {
        int ml = mel_len[lane];
        int cl = char_len[lane];
        v = alphas[(size_t)lane * T_ * L_ + (size_t)(ml - 1) * L_ + (cl - 1)];
    }
#pragma unroll
    for (int o = 16; o > 0; o >>= 1) v += __shfl_down(v, o, 32);
    if (lane == 0) *loss = -v / (float)B_;
}

extern "C" void kernel_launch(void* const* d_in, const int* in_sizes, int n_in,
                              void* d_out, int out_size, void* d_ws, size_t ws_size,
                              hipStream_t stream) {
    const float* mel          = (const float*)d_in[0];
    const float* mu_sigma     = (const float*)d_in[1];
    const int*   mel_lengths  = (const int*)d_in[2];
    const int*   char_lengths = (const int*)d_in[3];

    float* out    = (float*)d_out;
    float* logp   = out;                          // [B,T,L]
    float* lossp  = out + (size_t)B_ * T_ * L_;   // scalar
    float* alphas = lossp + 1;                    // [B,T,L]

    float* Bmat = (float*)d_ws;                   // [B, KP2, L, 2]
    float* At   = Bmat + (size_t)B_ * KP * L_;    // [B, KP2, T, 2]

    {
        int total = B_ * KP * L_;
        prep_B_kernel<<<(total + 255) / 256, 256, 0, stream>>>(mu_sigma, Bmat);
    }
    {
        int total = B_ * KP * T_;
        prep_At_kernel<<<(total + 255) / 256, 256, 0, stream>>>(mel, At);
    }
    {
        dim3 grid(T_ / 16, L_ / 32, B_);
        gemm_logprob_kernel<<<grid, 32, 0, stream>>>(At, Bmat, logp);
    }
    scan_alphas_kernel<<<B_, 32, 0, stream>>>(logp, alphas);
    loss_kernel<<<1, 32, 0, stream>>>(alphas, mel_lengths, char_lengths, lossp);
}